// NMS_34479997452785
// MI455X (gfx1250) — compile-verified
//
#include <hip/hip_runtime.h>
#include <hip/hip_bf16.h>
#include <stdint.h>

// NMS on 4096x4096 f32. 3 passes:
//  pass1: 3-tap separable triangle blur (WMMA f32 banded GEMM)  edge -> ws0
//  pass2: 9-tap separable triangle blur (WMMA f32 banded GEMM)  ws0 -> ws1
//  pass3: gradients + orientation + bilinear NMS                ws0,ws1 -> out,keep
// Requires ws_size >= 2 * 4096*4096 * 4 bytes (128 MB).

#define IMG_H 4096
#define IMG_W 4096

typedef __attribute__((ext_vector_type(2))) float v2f;
typedef __attribute__((ext_vector_type(8))) float v8f;

typedef __attribute__((address_space(3))) void lds_void_t;

__device__ __forceinline__ unsigned lds_byte_offset(void* p) {
  // generic -> LDS addrspace -> raw 32-bit LDS byte offset (what async-to-LDS wants)
  return (unsigned)(size_t)(lds_void_t*)p;
}

// Both filters are triangles: R=1 -> [1,2,1]/4, R=4 -> [1..5..1]/25.
// Coefficient of input position `pos` (tile-relative, may be <0 or >=16)
// contributing to output index m.
__device__ __forceinline__ float bcoef(int R, int m, int pos) {
  int d = pos - m;
  int ad = d < 0 ? -d : d;
  int w = (R + 1) - ad;
  float norm = 1.0f / (float)((R + 1) * (R + 1));
  return w > 0 ? (float)w * norm : 0.0f;
}

template <int R>
__global__ void __launch_bounds__(32)
conv_sep_wmma(const float* __restrict__ in, float* __restrict__ out) {
  // one wave per 16x16 output tile
  __shared__ float sIn[48][48];     // 3x3 tile neighborhood with zero pad
  __shared__ float sT[3][16][16];   // vertical-conv results (left/mid/right col tiles)

  const int tx = blockIdx.x, ty = blockIdx.y;
  const int lane = threadIdx.x;
  const int half = lane >> 4;       // 0: lanes 0-15, 1: lanes 16-31
  const int lo16 = lane & 15;

  // ---- stage 48x48 halo region via async global->LDS ----
  const int r0 = ty * 16 - 16;
  const int c0 = tx * 16 - 16;
  const bool interior = (tx > 0) && (tx < (int)gridDim.x - 1) &&
                        (ty > 0) && (ty < (int)gridDim.y - 1);
  if (interior) {
    // fully in-bounds: 48 rows x 12 16-byte chunks, all 16B aligned -> b128
    for (int i = lane; i < 48 * 12; i += 32) {
      int r = i / 12, cq = i % 12;
      unsigned ldsoff = lds_byte_offset(&sIn[r][cq * 4]);
      unsigned long long gaddr = (unsigned long long)(size_t)(
          in + (size_t)(r0 + r) * IMG_W + (c0 + cq * 4));
      asm volatile("global_load_async_to_lds_b128 %0, %1, off"
                   :: "v"(ldsoff), "v"(gaddr) : "memory");
    }
  } else {
    // border tiles: per-element with zero fill outside the image
    for (int i = lane; i < 48 * 48; i += 32) {
      int r = i / 48, c = i % 48;
      int gr = r0 + r, gc = c0 + c;
      if (gr >= 0 && gr < IMG_H && gc >= 0 && gc < IMG_W) {
        unsigned ldsoff = lds_byte_offset(&sIn[r][c]);
        unsigned long long gaddr =
            (unsigned long long)(size_t)(in + (size_t)gr * IMG_W + gc);
        asm volatile("global_load_async_to_lds_b32 %0, %1, off"
                     :: "v"(ldsoff), "v"(gaddr) : "memory");
      } else {
        sIn[r][c] = 0.0f;
      }
    }
  }
  asm volatile("s_wait_asynccnt 0x0" ::: "memory");

  // WMMA f32 16x16x4 operand lane mapping (wave32):
  //  A (16x4): a.x = A[m=lane&15][k = 2*(lane>>4)],     a.y = k+1
  //  B (4x16): b.x = B[k = 2*(lane>>4)][n = lane&15],   b.y = k+1
  //  C/D     : d[v] = D[m = v + 8*(lane>>4)][n = lane&15]
  const int kl0 = half * 2;
  const int kl1 = kl0 + 1;

  // Non-zero band-matrix K-chunks for a triangle filter with R<=4:
  //  prev tile -> chunk 3 only, mid tile -> chunks 0..3, next tile -> chunk 0.
  // Filter symmetry => the vertical A-operand chunks and horizontal B-operand
  // chunks are lane-for-lane identical, so compute the 6 band chunks once.
  const int TT[6] = {0, 1, 1, 1, 1, 2};  // neighbor tile (0=prev,1=mid,2=next)
  const int CK[6] = {3, 0, 1, 2, 3, 0};  // K chunk within that tile
  v2f band[6];
#pragma unroll
  for (int j = 0; j < 6; ++j) {
    int pos = (TT[j] - 1) * 16 + 4 * CK[j];
    band[j].x = bcoef(R, lo16, pos + kl0);
    band[j].y = bcoef(R, lo16, pos + kl1);
  }

  // ---- vertical pass: t_cc = Vband x in, for col tiles cc = 0,1,2 ----
  for (int cc = 0; cc < 3; ++cc) {
    v8f acc = {0.f, 0.f, 0.f, 0.f, 0.f, 0.f, 0.f, 0.f};
#pragma unroll
    for (int j = 0; j < 6; ++j) {
      int row = TT[j] * 16 + 4 * CK[j];
      v2f b;
      b.x = sIn[row + kl0][cc * 16 + lo16];
      b.y = sIn[row + kl1][cc * 16 + lo16];
      acc = __builtin_amdgcn_wmma_f32_16x16x4_f32(false, band[j], false, b,
                                                  (short)0, acc, false, false);
    }
#pragma unroll
    for (int v = 0; v < 8; ++v) sT[cc][v + half * 8][lo16] = acc[v];
  }

  // ---- horizontal pass: out = sum_cc t_cc x Hband ----
  v8f acc = {0.f, 0.f, 0.f, 0.f, 0.f, 0.f, 0.f, 0.f};
#pragma unroll
  for (int j = 0; j < 6; ++j) {
    int cc = TT[j];
    int col = 4 * CK[j];
    v2f a;
    a.x = sT[cc][lo16][col + kl0];
    a.y = sT[cc][lo16][col + kl1];
    acc = __builtin_amdgcn_wmma_f32_16x16x4_f32(false, a, false, band[j],
                                                (short)0, acc, false, false);
  }

#pragma unroll
  for (int v = 0; v < 8; ++v) {
    int gr = ty * 16 + v + half * 8;
    int gc = tx * 16 + lo16;
    out[(size_t)gr * IMG_W + gc] = acc[v];
  }
}

// ---------------- pass 3: gradients + orientation + bilinear NMS ----------------

__device__ __forceinline__ float A_(const float* a, int y, int x) {
  return a[(size_t)y * IMG_W + x];
}
__device__ __forceinline__ float gx_(const float* a, int y, int x) {
  if (x == 0) return A_(a, y, 1) - A_(a, y, 0);
  if (x == IMG_W - 1) return A_(a, y, IMG_W - 1) - A_(a, y, IMG_W - 2);
  return 0.5f * (A_(a, y, x + 1) - A_(a, y, x - 1));
}
__device__ __forceinline__ float gy_(const float* a, int y, int x) {
  if (y == 0) return A_(a, 1, x) - A_(a, 0, x);
  if (y == IMG_H - 1) return A_(a, IMG_H - 1, x) - A_(a, IMG_H - 2, x);
  return 0.5f * (A_(a, y + 1, x) - A_(a, y - 1, x));
}
__device__ __forceinline__ float gxx_(const float* a, int y, int x) {
  if (x == 0) return gx_(a, y, 1) - gx_(a, y, 0);
  if (x == IMG_W - 1) return gx_(a, y, IMG_W - 1) - gx_(a, y, IMG_W - 2);
  return 0.5f * (gx_(a, y, x + 1) - gx_(a, y, x - 1));
}
__device__ __forceinline__ float gxy_(const float* a, int y, int x) {
  if (x == 0) return gy_(a, y, 1) - gy_(a, y, 0);
  if (x == IMG_W - 1) return gy_(a, y, IMG_W - 1) - gy_(a, y, IMG_W - 2);
  return 0.5f * (gy_(a, y, x + 1) - gy_(a, y, x - 1));
}
__device__ __forceinline__ float gyy_(const float* a, int y, int x) {
  if (y == 0) return gy_(a, 1, x) - gy_(a, 0, x);
  if (y == IMG_H - 1) return gy_(a, IMG_H - 1, x) - gy_(a, IMG_H - 2, x);
  return 0.5f * (gy_(a, y + 1, x) - gy_(a, y - 1, x));
}

__device__ __forceinline__ float bilerp(const float* e, float x, float y) {
  int x0 = (int)x, y0 = (int)y;     // coords >= 0 after clip -> trunc == floor
  float dx0 = x - (float)x0, dy0 = y - (float)y0;
  float dx1 = 1.0f - dx0, dy1 = 1.0f - dy0;
  const float* r0 = e + (size_t)y0 * IMG_W;
  const float* r1 = r0 + IMG_W;
  return r0[x0] * dx1 * dy1 + r0[x0 + 1] * dx0 * dy1 +
         r1[x0] * dx1 * dy0 + r1[x0 + 1] * dx0 * dy0;
}

__global__ void __launch_bounds__(256)
nms_kernel(const float* __restrict__ sm, const float* __restrict__ a,
           float* __restrict__ out, int* __restrict__ keep) {
  int idx = blockIdx.x * blockDim.x + threadIdx.x;
  if (idx >= IMG_H * IMG_W) return;
  int y = idx / IMG_W, x = idx - y * IMG_W;

  float oxx = gxx_(a, y, x);
  float oyy = gyy_(a, y, x);
  float oxy = gxy_(a, y, x);

  float noxy = -oxy;
  float sgn = (noxy > 0.0f) ? 1.0f : ((noxy < 0.0f) ? -1.0f : 0.0f);
  const float PI_F = 3.14159265358979323846f;
  float t = atanf(oyy * sgn / (oxx + 1e-5f));
  float ori = t - floorf(t / PI_F) * PI_F;  // jnp.remainder(t, pi)

  float cs = cosf(ori), sn = sinf(ori);
  float xf = (float)x, yf = (float)y;
  const float xmax = (float)IMG_W - 1.001f;
  const float ymax = (float)IMG_H - 1.001f;

  auto cx = [&](float v) { return fminf(fmaxf(v, 0.0f), xmax); };
  auto cy = [&](float v) { return fminf(fmaxf(v, 0.0f), ymax); };

  float e01 = bilerp(sm, cx(2.0f * cs + xf), cy(2.0f * sn + yf));
  float e02 = bilerp(sm, cx(cs + xf),        cy(sn + yf));
  float e03 = bilerp(sm, cx(-2.0f * cs + xf), cy(-2.0f * sn + yf));
  float e04 = bilerp(sm, cx(-cs + xf),       cy(-sn + yf));

  float ev = sm[(size_t)y * IMG_W + x];
  float e = ev * 1.01f;
  int k = ((e < e01) || (e < e02) || (e < e03) || (e < e04)) ? 0 : 1;
  out[idx] = ev * (float)k;
  keep[idx] = k;
}

extern "C" void kernel_launch(void* const* d_in, const int* in_sizes, int n_in,
                              void* d_out, int out_size, void* d_ws, size_t ws_size,
                              hipStream_t stream) {
  (void)in_sizes; (void)n_in; (void)out_size; (void)ws_size;
  const size_t N = (size_t)IMG_H * IMG_W;

  const float* edge = (const float*)d_in[0];
  float* sm = (float*)d_ws;            // smoothed edge (pass1 out), 64 MB
  float* av = sm + N;                  // heavier-smoothed 'a' (pass2 out), 64 MB
  float* out = (float*)d_out;          // output 0: out (float), N elems
  int* keep = (int*)d_out + N;         // output 1: keep (int32), N elems

  dim3 gridC(IMG_W / 16, IMG_H / 16);  // one wave32 per 16x16 tile
  conv_sep_wmma<1><<<gridC, 32, 0, stream>>>(edge, sm);
  conv_sep_wmma<4><<<gridC, 32, 0, stream>>>(sm, av);

  int n = IMG_H * IMG_W;
  nms_kernel<<<(n + 255) / 256, 256, 0, stream>>>(sm, av, out, keep);
}